// GaussianCoxPHDecayLossTraditional_16501264351666
// MI455X (gfx1250) — compile-verified
//
#include <hip/hip_runtime.h>
#include <hip/hip_bf16.h>
#include <math.h>

#define NN 2048
#define MM 32
#define NWAVES ((NN / 128) * MM * 8)   // 16 * 32 * 8 = 4096 partials

typedef float v2f  __attribute__((ext_vector_type(2)));
typedef float v8f  __attribute__((ext_vector_type(8)));
typedef unsigned int u32x4 __attribute__((ext_vector_type(4)));
typedef int  i32x4 __attribute__((ext_vector_type(4)));
typedef int  i32x8 __attribute__((ext_vector_type(8)));

// ---------------------------------------------------------------------------
// Kernel 1: per-column max of (expressions - predicted_tr)  -> mx[32]
// ---------------------------------------------------------------------------
__global__ void cox_colmax_kernel(const float* __restrict__ expr,
                                  const float* __restrict__ tr,
                                  float* __restrict__ mx) {
    __shared__ float sred[256];
    const int k = blockIdx.x;
    float m = -3.4028235e38f;
    for (int a = threadIdx.x; a < NN; a += 256) {
        const int idx = a * MM + k;
        m = fmaxf(m, expr[idx] - tr[idx]);
    }
    sred[threadIdx.x] = m;
    __syncthreads();
    for (int s = 128; s > 0; s >>= 1) {
        if (threadIdx.x < s)
            sred[threadIdx.x] = fmaxf(sred[threadIdx.x], sred[threadIdx.x + s]);
        __syncthreads();
    }
    if (threadIdx.x == 0) mx[k] = sred[0];
}

// ---------------------------------------------------------------------------
// Kernel 2: ti_T[k,a] = exp((e-t) - mx[k]) / (2*sigma*sqrt(2))
//           eh_T[k,a] = exp(risk[a,k])          (transposed, contiguous cols)
// ---------------------------------------------------------------------------
__global__ void cox_transform_kernel(const float* __restrict__ expr,
                                     const float* __restrict__ tr,
                                     const float* __restrict__ risk,
                                     const float* __restrict__ sigma,
                                     const float* __restrict__ mx,
                                     float* __restrict__ tiT,
                                     float* __restrict__ ehT) {
    const int idx = blockIdx.x * 256 + threadIdx.x;
    if (idx >= NN * MM) return;
    const int a = idx >> 5;      // row
    const int k = idx & 31;      // column
    const float inv = 1.0f / (2.0f * sigma[0] * 1.41421356237309505f);
    const float x = expr[idx] - tr[idx];
    tiT[k * NN + a] = expf(x - mx[k]) * inv;
    ehT[k * NN + a] = expf(risk[idx]);
}

// ---------------------------------------------------------------------------
// Kernel 3: main pairwise kernel. Block = 256 threads (8 waves), fixed k.
// Each wave owns 16 j rows; a-loop in chunks of 4 folded into
// V_WMMA_F32_16X16X4_F32 with broadcast-B (B[K][n] = expH[a0+K] for all n).
// Per-wave partial sums go to a fixed workspace slot (deterministic).
// ---------------------------------------------------------------------------
__global__ void __launch_bounds__(256)
cox_main_kernel(const float* __restrict__ tiT,
                const float* __restrict__ ehT,
                const float* __restrict__ risk,
                const float* __restrict__ event,
                float* __restrict__ partial) {
    __shared__ __align__(16) float lds_ti[NN];
    __shared__ __align__(16) float lds_eh[NN];

    const int k = blockIdx.y;

    // --- Stage both 8KB columns into LDS with the Tensor Data Mover ---
    if (threadIdx.x < 32u) {
        const unsigned long long gti =
            (unsigned long long)(uintptr_t)(tiT + (size_t)k * NN);
        const unsigned long long geh =
            (unsigned long long)(uintptr_t)(ehT + (size_t)k * NN);
        const unsigned int lti = (unsigned int)(uintptr_t)&lds_ti[0];
        const unsigned int leh = (unsigned int)(uintptr_t)&lds_eh[0];

        // D# group 1 (shared by both loads): data_size=4B, 1-D tile of 2048.
        i32x8 g1;
        g1[0] = 0x00020000;                 // wg_mask=0, data_size=2 (4 bytes)
        g1[1] = (int)((NN & 0xFFFF) << 16); // tensor_dim0 low16 (bits 63:48)
        g1[2] = (int)(1 << 16);             // tensor_dim0 hi=0; tensor_dim1=1
        g1[3] = (int)(NN << 16);            // tensor_dim1 hi=0; tile_dim0=2048
        g1[4] = 0;                          // tile_dim1=0 (unused), tile_dim2=0
        g1[5] = NN;                         // tensor_dim0_stride low32
        g1[6] = 0;
        g1[7] = 0;
        const i32x4 z4 = {0, 0, 0, 0};
        const i32x8 z8 = {0, 0, 0, 0, 0, 0, 0, 0};

        u32x4 g0a;
        g0a[0] = 1u;                                     // count=1
        g0a[1] = lti;                                    // lds_addr
        g0a[2] = (unsigned int)(gti & 0xFFFFFFFFull);    // global_addr lo
        g0a[3] = (unsigned int)((gti >> 32) & 0x01FFFFFFull) | 0x80000000u; // type=2

        u32x4 g0b;
        g0b[0] = 1u;
        g0b[1] = leh;
        g0b[2] = (unsigned int)(geh & 0xFFFFFFFFull);
        g0b[3] = (unsigned int)((geh >> 32) & 0x01FFFFFFull) | 0x80000000u;

        __builtin_amdgcn_tensor_load_to_lds(g0a, g1, z4, z4, z8, 0);
        __builtin_amdgcn_tensor_load_to_lds(g0b, g1, z4, z4, z8, 0);
        __builtin_amdgcn_s_wait_tensorcnt(0);
    }
    __syncthreads();

    const int lane = threadIdx.x & 31;
    const int wave = threadIdx.x >> 5;
    const int jb   = blockIdx.x * 128 + wave * 16;   // wave's 16-row j block
    const int jl   = lane & 15;                      // A-matrix M = lane
    const int koff = (lane >> 4) << 1;               // A/B K slots: {0,1} or {2,3}

    const float tij = lds_ti[jb + jl];

    v8f c = {0.f, 0.f, 0.f, 0.f, 0.f, 0.f, 0.f, 0.f};
    for (int a0 = 0; a0 < NN; a0 += 4) {
        const float2 tt = *(const float2*)&lds_ti[a0 + koff];
        const float2 ee = *(const float2*)&lds_eh[a0 + koff];
        // phi(j,a) = 0.5*(1+erf(ti[a]-ti[j]))
        v2f A = {0.5f * (1.0f + erff(tt.x - tij)),
                 0.5f * (1.0f + erff(tt.y - tij))};
        v2f B = {ee.x, ee.y};   // broadcast across all 16 N columns
        c = __builtin_amdgcn_wmma_f32_16x16x4_f32(
            /*neg_a=*/false, A, /*neg_b=*/false, B,
            /*c_mod=*/(short)0, c, /*reuse_a=*/false, /*reuse_b=*/false);
    }

    // Epilogue: lanes 0-15 hold S[jb+0..7] in c[0..7]; lanes 16-31 hold jb+8..15.
    float wsum = 0.0f;
#pragma unroll
    for (int r = 0; r < 8; ++r) {
        const int jloc = (lane < 16) ? r : (r + 8);
        const int j2   = jb + jloc;
        const float S  = c[r] + 0.5f * lds_eh[j2];   // diagonal +0.5 correction
        const float contrib =
            (risk[j2 * MM + k] - logf(S)) * event[j2 * MM + k];
        wsum += __shfl(contrib, 0, 32) + __shfl(contrib, 16, 32);
    }
    if (lane == 0) {
        const int gwave = (blockIdx.y * gridDim.x + blockIdx.x) * 8 + wave;
        partial[gwave] = wsum;
    }
}

// ---------------------------------------------------------------------------
// Kernel 4: deterministic final reduction of the 4096 per-wave partials.
// Fixed serial order per thread + fixed LDS tree -> bitwise reproducible.
// ---------------------------------------------------------------------------
__global__ void cox_reduce_kernel(const float* __restrict__ partial,
                                  float* __restrict__ out) {
    __shared__ float sred[256];
    float t = 0.0f;
    for (int i = threadIdx.x; i < NWAVES; i += 256) t += partial[i];
    sred[threadIdx.x] = t;
    __syncthreads();
    for (int s = 128; s > 0; s >>= 1) {
        if (threadIdx.x < s) sred[threadIdx.x] += sred[threadIdx.x + s];
        __syncthreads();
    }
    if (threadIdx.x == 0) out[0] = -sred[0];
}

// ---------------------------------------------------------------------------
extern "C" void kernel_launch(void* const* d_in, const int* in_sizes, int n_in,
                              void* d_out, int out_size, void* d_ws, size_t ws_size,
                              hipStream_t stream) {
    const float* risk  = (const float*)d_in[0];  // predicted_risk
    const float* expr  = (const float*)d_in[1];  // expressions
    const float* tr    = (const float*)d_in[2];  // predicted_tr
    const float* event = (const float*)d_in[3];  // event_indicator
    const float* sigma = (const float*)d_in[4];  // sigma (1 element)

    float* mx      = (float*)d_ws;        // 32 floats
    float* tiT     = mx + 32;             // 2048*32 floats (transposed)
    float* ehT     = tiT + NN * MM;       // 2048*32 floats (transposed)
    float* partial = ehT + NN * MM;       // 4096 per-wave partials
    float* out     = (float*)d_out;

    cox_colmax_kernel<<<dim3(MM), dim3(256), 0, stream>>>(expr, tr, mx);
    cox_transform_kernel<<<dim3((NN * MM) / 256), dim3(256), 0, stream>>>(
        expr, tr, risk, sigma, mx, tiT, ehT);
    cox_main_kernel<<<dim3(NN / 128, MM), dim3(256), 0, stream>>>(
        tiT, ehT, risk, event, partial);
    cox_reduce_kernel<<<dim3(1), dim3(256), 0, stream>>>(partial, out);
}